// SSM_63350767616145
// MI455X (gfx1250) — compile-verified
//
#include <hip/hip_runtime.h>
#include <hip/hip_bf16.h>
#include <math.h>

// ---------------- problem constants ----------------
#define B_SZ    2
#define SEQ_L   1024
#define D_INNER 2048
#define DT_RANK 64
#define D_STATE 16
#define KPROJ   (DT_RANK + 2 * D_STATE)   // 96
#define MROWS   (B_SZ * SEQ_L)            // 2048

typedef __attribute__((ext_vector_type(2))) float v2f;
typedef __attribute__((ext_vector_type(8))) float v8f;

// Generic pointers into LDS carry the LDS byte offset in their low 32 bits
// (ISA 10.2: only the low 32 address bits are used), which is what the async-load
// VDST operand wants.
__device__ __forceinline__ unsigned lds_off(const void* p) {
  return (unsigned)(uintptr_t)p;
}

// Async copy 16 bytes global -> LDS (GLOBAL_LOAD_ASYNC_TO_LDS_B128, ASYNCcnt).
__device__ __forceinline__ void async_ld_b128(unsigned ldsaddr, const void* gptr) {
  asm volatile("global_load_async_to_lds_b128 %0, %1, off"
               :: "v"(ldsaddr), "v"((unsigned long long)(uintptr_t)gptr)
               : "memory");
}
__device__ __forceinline__ void wait_asynccnt0() {
  asm volatile("s_wait_asynccnt 0x0" ::: "memory");
}

// --------------------------------------------------------------------------
// Kernel 1: x_dbl[m, k] = sum_d x[m, d] * W_xproj[k, d]
// M = 2048, N = 96, K = 2048, fp32 WMMA 16x16x4.
// One wave per 16-row M tile; computes all 6 N tiles so the A fragment is
// loaded once per K step. A-frag layout (ISA 7.12.2, 32-bit A 16x4):
//   lane = m%16 (halves replicated), VGPR j + 2*(lane/16) = k offset.
// B/C/D: lane%16 = column, VGPR r + 8*(lane/16) = row.
// --------------------------------------------------------------------------
__global__ __launch_bounds__(32) void gemm_xproj_wmma(
    const float* __restrict__ x,      // (MROWS, D_INNER)
    const float* __restrict__ Wx,     // (KPROJ, D_INNER)
    float* __restrict__ xdbl)         // (MROWS, KPROJ)
{
  const int lane = threadIdx.x;
  const int m0   = blockIdx.x * 16;
  const int ml   = lane & 15;
  const int kh   = lane >> 4;        // 0 or 1: which K half-pair

  v8f acc[6];
#pragma unroll
  for (int t = 0; t < 6; ++t) acc[t] = (v8f)0.0f;

  const float* arow = x + (size_t)(m0 + ml) * D_INNER + 2 * kh;

  for (int c0 = 0; c0 < D_INNER; c0 += 4) {
    v2f a = *(const v2f*)(arow + c0);
#pragma unroll
    for (int t = 0; t < 6; ++t) {
      const int n = t * 16 + ml;     // output column (k index of x_dbl)
      v2f b = *(const v2f*)(Wx + (size_t)n * D_INNER + c0 + 2 * kh);
      acc[t] = __builtin_amdgcn_wmma_f32_16x16x4_f32(
          false, a, false, b, (short)0, acc[t], false, false);
    }
  }

#pragma unroll
  for (int t = 0; t < 6; ++t) {
    const int col = t * 16 + ml;
#pragma unroll
    for (int r = 0; r < 8; ++r) {
      const int row = m0 + r + 8 * kh;
      xdbl[(size_t)row * KPROJ + col] = acc[t][r];
    }
  }
}

// --------------------------------------------------------------------------
// Kernel 2: delta[m, d] = softplus( sum_r x_dbl[m, r] * W_dt[d, r] + b_dt[d] )
// M = 2048, N = 2048, K = 64. One wave: 16 rows x 64 cols (4 WMMA tiles).
// --------------------------------------------------------------------------
__global__ __launch_bounds__(32) void gemm_dt_softplus_wmma(
    const float* __restrict__ xdbl,   // (MROWS, KPROJ) : use cols [0,64)
    const float* __restrict__ Wdt,    // (D_INNER, DT_RANK)
    const float* __restrict__ bdt,    // (D_INNER)
    float* __restrict__ delta)        // (MROWS, D_INNER)
{
  const int lane = threadIdx.x;
  const int m0   = blockIdx.x * 16;
  const int n0   = blockIdx.y * 64;
  const int ml   = lane & 15;
  const int kh   = lane >> 4;

  v8f acc[4];
#pragma unroll
  for (int t = 0; t < 4; ++t) acc[t] = (v8f)0.0f;

  const float* arow = xdbl + (size_t)(m0 + ml) * KPROJ + 2 * kh;

#pragma unroll
  for (int r0 = 0; r0 < DT_RANK; r0 += 4) {
    v2f a = *(const v2f*)(arow + r0);
#pragma unroll
    for (int t = 0; t < 4; ++t) {
      const int n = n0 + t * 16 + ml;
      v2f b = *(const v2f*)(Wdt + (size_t)n * DT_RANK + r0 + 2 * kh);
      acc[t] = __builtin_amdgcn_wmma_f32_16x16x4_f32(
          false, a, false, b, (short)0, acc[t], false, false);
    }
  }

#pragma unroll
  for (int t = 0; t < 4; ++t) {
    const int col  = n0 + t * 16 + ml;
    const float bb = bdt[col];
#pragma unroll
    for (int r = 0; r < 8; ++r) {
      const int row = m0 + r + 8 * kh;
      float z  = acc[t][r] + bb;
      float sp = (z > 20.0f) ? z : log1pf(__expf(z));   // softplus
      delta[(size_t)row * D_INNER + col] = sp;
    }
  }
}

// --------------------------------------------------------------------------
// Kernel 3: selective scan. Lane-per-(b, d, n): 64K threads = 2048 wave32s.
//   h[n] <- exp(delta * A[d,n]) * h[n] + delta * x * B[l,n]
//   y     = sum_n h[n] * C[l,n] + D[d] * x        (16-lane shfl_xor tree)
// Per-step operands staged through LDS in 64-step chunks (16 KB / block)
// using GLOBAL_LOAD_ASYNC_TO_LDS_B128: each staging array is 64x16 floats
// = 256 threads x one 16B transfer (all addresses 16B aligned).
// Block = 256 threads = 16 (b,d) rows x 16 state channels.
// --------------------------------------------------------------------------
#define CHUNK 64

__global__ __launch_bounds__(256) void ssm_scan(
    const float* __restrict__ x,      // (MROWS, D_INNER)
    const float* __restrict__ A_log,  // (D_INNER, D_STATE)
    const float* __restrict__ Dvec,   // (D_INNER)
    const float* __restrict__ xdbl,   // (MROWS, KPROJ): B at [64,80), C at [80,96)
    const float* __restrict__ delta,  // (MROWS, D_INNER)
    float* __restrict__ y)            // (MROWS, D_INNER)
{
  __shared__ float sDelta[CHUNK * 16];
  __shared__ float sX    [CHUNK * 16];
  __shared__ float sB    [CHUNK * 16];
  __shared__ float sC    [CHUNK * 16];

  const int b   = blockIdx.x >> 7;             // 0..1
  const int d0  = (blockIdx.x & 127) * 16;     // 16 channels per block
  const int row = threadIdx.x >> 4;            // which d within block
  const int n   = threadIdx.x & 15;            // state channel
  const int d   = d0 + row;

  // This thread's 16B slice of each staging array (one float4 per array).
  const int e4 = threadIdx.x * 4;              // element index of first float
  const int si = e4 >> 4;                      // chunk-step index  (0..63)
  const int sj = e4 & 15;                      // column within row (0,4,8,12)
  const unsigned ldsDelta = lds_off(&sDelta[e4]);
  const unsigned ldsX     = lds_off(&sX[e4]);
  const unsigned ldsB     = lds_off(&sB[e4]);
  const unsigned ldsC     = lds_off(&sC[e4]);

  const float an = -__expf(A_log[(size_t)d * D_STATE + n]);
  const float Dd = Dvec[d];
  float h = 0.0f;

  for (int lc = 0; lc < SEQ_L; lc += CHUNK) {
    const size_t mbase = (size_t)b * SEQ_L + lc;
    const size_t mrow  = mbase + si;

    // async chunk stage: 4 x b128 per thread, straight into LDS
    async_ld_b128(ldsDelta, &delta[mrow * D_INNER + d0 + sj]);
    async_ld_b128(ldsX,     &x    [mrow * D_INNER + d0 + sj]);
    async_ld_b128(ldsB,     &xdbl [mrow * KPROJ + DT_RANK + sj]);
    async_ld_b128(ldsC,     &xdbl [mrow * KPROJ + DT_RANK + D_STATE + sj]);
    wait_asynccnt0();
    __syncthreads();

    // prefetch next chunk's lines into cache while this chunk computes
    if (lc + CHUNK < SEQ_L) {
      const size_t mnext = mrow + CHUNK;
      __builtin_prefetch(&delta[mnext * D_INNER + d0 + sj], 0, 3);
      __builtin_prefetch(&x    [mnext * D_INNER + d0 + sj], 0, 3);
      __builtin_prefetch(&xdbl [mnext * KPROJ + DT_RANK + sj], 0, 3);
    }

    for (int i = 0; i < CHUNK; ++i) {
      const float dlt = sDelta[i * 16 + row];   // broadcast across 16 lanes
      const float xv  = sX    [i * 16 + row];
      const float Bn  = sB    [i * 16 + n];
      const float Cn  = sC    [i * 16 + n];

      const float dA = __expf(dlt * an);
      h = dA * h + (dlt * xv) * Bn;

      float p = h * Cn;
      // reduce across the 16 state lanes (stays inside the group of 16)
      p += __shfl_xor(p, 1);
      p += __shfl_xor(p, 2);
      p += __shfl_xor(p, 4);
      p += __shfl_xor(p, 8);

      if (n == 0) {
        const size_t m = mbase + i;
        y[m * D_INNER + d] = p + Dd * xv;
      }
    }
    __syncthreads();
  }
}

// --------------------------------------------------------------------------
extern "C" void kernel_launch(void* const* d_in, const int* in_sizes, int n_in,
                              void* d_out, int out_size, void* d_ws, size_t ws_size,
                              hipStream_t stream) {
  const float* x     = (const float*)d_in[0];   // (2,1024,2048)
  const float* Wx    = (const float*)d_in[1];   // (96,2048)
  const float* Wdt   = (const float*)d_in[2];   // (2048,64)
  const float* bdt   = (const float*)d_in[3];   // (2048)
  const float* A_log = (const float*)d_in[4];   // (2048,16)
  const float* Dv    = (const float*)d_in[5];   // (2048)
  float* yout = (float*)d_out;

  float* xdbl  = (float*)d_ws;                                    // 2048*96  f32
  float* delta = (float*)((char*)d_ws +
                          (size_t)MROWS * KPROJ * sizeof(float)); // 2048*2048 f32

  gemm_xproj_wmma<<<dim3(MROWS / 16), dim3(32), 0, stream>>>(x, Wx, xdbl);
  gemm_dt_softplus_wmma<<<dim3(MROWS / 16, D_INNER / 64), dim3(32), 0, stream>>>(
      xdbl, Wdt, bdt, delta);
  ssm_scan<<<dim3(B_SZ * D_INNER / 16), dim3(256), 0, stream>>>(
      x, A_log, Dv, xdbl, delta, yout);
}